// globalGNN_46222438039626
// MI455X (gfx1250) — compile-verified
//
#include <hip/hip_runtime.h>
#include <math.h>

// ---- problem constants (from reference) ----
#define N_NODES  50000
#define N_EDGES  800000
#define HID      128
#define N_GRAPHS 64
#define ROW_TILES (N_NODES / 16)   // 3125, exact

typedef float v2f __attribute__((ext_vector_type(2)));
typedef float v8f __attribute__((ext_vector_type(8)));

// ---------------------------------------------------------------------------
// dst[] = src[]  (float4 vectorized, grid-stride)
// ---------------------------------------------------------------------------
__global__ void copy_f4(float4* __restrict__ dst, const float4* __restrict__ src, int n4) {
    int i = blockIdx.x * blockDim.x + threadIdx.x;
    int stride = gridDim.x * blockDim.x;
    for (; i < n4; i += stride) dst[i] = src[i];
}

// ---------------------------------------------------------------------------
// agg[dst[e], :] += x[src[e], :]   (one wave per edge, 4 floats/lane)
// agg must already contain the self term (copy of x).
// ---------------------------------------------------------------------------
__global__ __launch_bounds__(256) void scatter_add(
    float* __restrict__ agg, const float* __restrict__ x,
    const int* __restrict__ src, const int* __restrict__ dst) {
    int t    = blockIdx.x * blockDim.x + threadIdx.x;
    int e    = t >> 5;
    int lane = t & 31;
    if (e >= N_EDGES) return;
    int s = src[e];
    int d = dst[e];
    const float4 v = *(const float4*)(x + (size_t)s * HID + lane * 4);
    float* p = agg + (size_t)d * HID + lane * 4;
    atomicAdd(p + 0, v.x);
    atomicAdd(p + 1, v.y);
    atomicAdd(p + 2, v.z);
    atomicAdd(p + 3, v.w);
}

// ---------------------------------------------------------------------------
// out = gelu(in @ W + bias), in/out: [N_NODES,128] row-major, W: [128,128]
// row-major (in-dim major).  One wave owns a 16-column tile; B fragments for
// all of K=128 live in registers; grid-stride over 16-row tiles.
//
// V_WMMA_F32_16X16X4_F32 operand layouts (wave32):
//   A 16x4 : lane&15 = M; VGPR0/1 = {K, K+1} with K = 2*(lane>>4)  -> float2
//   B 4x16 : lane&15 = N; VGPR0/1 = rows {K, K+1}, K = 2*(lane>>4)
//   C/D    : lane&15 = N; VGPR i -> M = i + 8*(lane>>4)
// ---------------------------------------------------------------------------
__global__ __launch_bounds__(256) void gemm_gelu(
    float* __restrict__ out, const float* __restrict__ in,
    const float* __restrict__ W, const float* __restrict__ bias) {
    const int tid  = threadIdx.x;
    const int wave = tid >> 5;          // 0..7 -> column tile
    const int lane = tid & 31;
    const int m    = lane & 15;
    const int hi   = lane >> 4;
    const int col  = wave * 16 + m;     // 0..127

    // Preload this wave's B fragments for the whole K=128 (32 steps of K=4).
    v2f b[32];
#pragma unroll
    for (int s = 0; s < 32; ++s) {
        const int k = 4 * s + 2 * hi;
        b[s].x = W[(k + 0) * HID + col];
        b[s].y = W[(k + 1) * HID + col];
    }
    const float bv = bias[col];

    for (int t = blockIdx.x; t < ROW_TILES; t += gridDim.x) {
        const float* arow = in + (size_t)(t * 16 + m) * HID + 2 * hi;
        v8f c = {};
#pragma unroll
        for (int s = 0; s < 32; ++s) {
            v2f a = *(const v2f*)(arow + 4 * s);
            c = __builtin_amdgcn_wmma_f32_16x16x4_f32(
                    /*neg_a=*/false, a, /*neg_b=*/false, b[s],
                    /*c_mod=*/(short)0, c, /*reuse_a=*/false, /*reuse_b=*/false);
        }
        float* orow = out + (size_t)(t * 16 + 8 * hi) * HID + col;
#pragma unroll
        for (int i = 0; i < 8; ++i) {
            float v = c[i] + bv;
            v = 0.5f * v * (1.0f + erff(v * 0.7071067811865475f));  // exact GELU
            orow[i * HID] = v;
        }
    }
}

// ---------------------------------------------------------------------------
// out[g] = sum over nodes n with batch[n]==g of (h[n,:]·Wfc + bfc)
// one wave per node; shuffle reduce; lane0 atomicAdd.
// ---------------------------------------------------------------------------
__global__ void zero_out(float* out) {
    if (threadIdx.x < N_GRAPHS) out[threadIdx.x] = 0.0f;
}

__global__ __launch_bounds__(256) void readout(
    float* __restrict__ out, const float* __restrict__ h,
    const float* __restrict__ Wfc, const float* __restrict__ bfc,
    const int* __restrict__ batch) {
    int t    = blockIdx.x * blockDim.x + threadIdx.x;
    int n    = t >> 5;
    int lane = t & 31;
    if (n >= N_NODES) return;
    float4 hv = *(const float4*)(h + (size_t)n * HID + lane * 4);
    float4 wv = *(const float4*)(Wfc + lane * 4);
    float v = hv.x * wv.x + hv.y * wv.y + hv.z * wv.z + hv.w * wv.w;
#pragma unroll
    for (int off = 16; off > 0; off >>= 1) v += __shfl_down(v, off, 32);
    if (lane == 0) atomicAdd(out + batch[n], v + bfc[0]);
}

// ---------------------------------------------------------------------------
extern "C" void kernel_launch(void* const* d_in, const int* in_sizes, int n_in,
                              void* d_out, int out_size, void* d_ws, size_t ws_size,
                              hipStream_t stream) {
    const float* x     = (const float*)d_in[0];
    const int*   ei    = (const int*)d_in[1];
    const int*   batch = (const int*)d_in[2];
    const float* W1a   = (const float*)d_in[3];
    const float* b1a   = (const float*)d_in[4];
    const float* W2a   = (const float*)d_in[5];
    const float* b2a   = (const float*)d_in[6];
    const float* W1b   = (const float*)d_in[7];
    const float* b1b   = (const float*)d_in[8];
    const float* W2b   = (const float*)d_in[9];
    const float* b2b   = (const float*)d_in[10];
    const float* Wfc   = (const float*)d_in[11];
    const float* bfc   = (const float*)d_in[12];

    const int* src = ei;             // edge_index[0, :]
    const int* dst = ei + N_EDGES;   // edge_index[1, :]

    float* bufA = (float*)d_ws;                       // 25.6 MB
    float* bufB = bufA + (size_t)N_NODES * HID;       // 25.6 MB

    const int NFEAT4   = N_NODES * HID / 4;           // 1.6M float4
    const int CP_BLKS  = 2048;
    const int SC_BLKS  = (N_EDGES * 32) / 256;        // 100000 exact
    const int GM_BLKS  = 1024;                        // grid-stride over 3125 tiles
    const int RO_BLKS  = (N_NODES * 32 + 255) / 256;  // 6250

    float* out = (float*)d_out;

    // ---- layer 1: agg = x + scatter; h = gelu(gelu(agg W1a+b1a) W2a+b2a) ----
    copy_f4<<<CP_BLKS, 256, 0, stream>>>((float4*)bufA, (const float4*)x, NFEAT4);
    scatter_add<<<SC_BLKS, 256, 0, stream>>>(bufA, x, src, dst);
    gemm_gelu<<<GM_BLKS, 256, 0, stream>>>(bufB, bufA, W1a, b1a);
    gemm_gelu<<<GM_BLKS, 256, 0, stream>>>(bufA, bufB, W2a, b2a);  // bufA = h1

    // ---- layer 2 ----
    copy_f4<<<CP_BLKS, 256, 0, stream>>>((float4*)bufB, (const float4*)bufA, NFEAT4);
    scatter_add<<<SC_BLKS, 256, 0, stream>>>(bufB, bufA, src, dst);
    gemm_gelu<<<GM_BLKS, 256, 0, stream>>>(bufA, bufB, W1b, b1b);
    gemm_gelu<<<GM_BLKS, 256, 0, stream>>>(bufB, bufA, W2b, b2b);  // bufB = h2

    // ---- readout ----
    zero_out<<<1, 64, 0, stream>>>(out);
    readout<<<RO_BLKS, 256, 0, stream>>>(out, bufB, Wfc, bfc, batch);
}